// motion_Dynamic_filter_67946382622953
// MI455X (gfx1250) — compile-verified
//
#include <hip/hip_runtime.h>

typedef float v4f __attribute__((ext_vector_type(4)));

constexpr int N = 8, C = 64, H = 128, W = 128;
constexpr int TILE_H = 8;
constexpr int LDS_ROWS = TILE_H + 2;          // rows h0-1 .. h0+TILE_H
constexpr int LDS_STRIDE = 4 + W + 4;         // 136 floats; 136*4B = 544 = 16B-aligned rows
constexpr int LDS_FLOATS = LDS_ROWS * LDS_STRIDE;   // 1360
constexpr float SLOPE = 0.2f;

__global__ __launch_bounds__(256) void fac_leaky_kernel(
    const float* __restrict__ feat,
    const float* __restrict__ flt,
    float* __restrict__ out)
{
    __shared__ __align__(16) float smem[LDS_FLOATS];

    const int tid = threadIdx.x;
    const int b   = blockIdx.x;                 // N*C*(H/TILE_H) blocks
    const int ht  = b & 15;                     // H/TILE_H = 16
    const int c   = (b >> 4) & 63;
    const int n   = b >> 10;
    const int h0  = ht * TILE_H;

    // ---- 1) zero the LDS tile with b128 stores (border halo + pad columns) ----
    v4f* smem4 = (v4f*)smem;
#pragma unroll
    for (int i = tid; i < LDS_FLOATS / 4; i += 256)   // 340 v4f slots: 2 iterations max
        smem4[i] = (v4f)0.0f;
    __syncthreads();

    // ---- 2) async-stage feat rows h0-1 .. h0+TILE_H into LDS (ASYNCcnt path) ----
    const float* featc = feat + (n * C + c) * (H * W);
    for (int j = tid; j < LDS_ROWS * (W / 4); j += 256) {
        const int r  = j >> 5;                  // LDS row 0..9  (W/4 = 32)
        const int cw = j & 31;                  // float4 column
        const int gh = h0 - 1 + r;              // global feat row
        if (gh >= 0 && gh < H) {                // EXEC-masked lanes skip the transfer
            unsigned lds_addr = (unsigned)(uintptr_t)&smem[r * LDS_STRIDE + 4 + cw * 4];
            const float* gaddr = featc + gh * W + cw * 4;
            asm volatile("global_load_async_to_lds_b128 %0, %1, off"
                         :: "v"(lds_addr), "v"(gaddr)
                         : "memory");
        }
    }
    asm volatile("s_wait_asynccnt 0x0" ::: "memory");
    __syncthreads();

    // ---- 3) compute: one wave per output row, 4 pixels per lane ----
    const int tr    = tid >> 5;                 // tile row 0..7
    const int w4    = tid & 31;                 // float4 group along W
    const int h     = h0 + tr;
    const int wbase = w4 * 4;

    // 9 per-pixel filter taps: used exactly once -> non-temporal b128 streams
    const float* fptr = flt + ((n * C + c) * 9 * H + h) * W + wbase;
    v4f fw[9];
#pragma unroll
    for (int t = 0; t < 9; ++t)
        fw[t] = __builtin_nontemporal_load((const v4f*)(fptr + t * (H * W)));

    float acc0 = 0.f, acc1 = 0.f, acc2 = 0.f, acc3 = 0.f;
#pragma unroll
    for (int dh = 0; dh < 3; ++dh) {
        const float* row = &smem[(tr + dh) * LDS_STRIDE + 4 + wbase];
        const float  l   = row[-1];             // data col wbase-1 (0 at border)
        const v4f    cv  = *(const v4f*)row;    // 16B-aligned ds_load_b128
        const float  rgt = row[4];              // data col wbase+4 (0 at border)
        const float v[6] = { l, cv.x, cv.y, cv.z, cv.w, rgt };
#pragma unroll
        for (int dw = 0; dw < 3; ++dw) {
            const v4f f = fw[dh * 3 + dw];
            acc0 = fmaf(v[0 + dw], f.x, acc0);
            acc1 = fmaf(v[1 + dw], f.y, acc1);
            acc2 = fmaf(v[2 + dw], f.z, acc2);
            acc3 = fmaf(v[3 + dw], f.w, acc3);
        }
    }

    v4f o;
    o.x = acc0 >= 0.f ? acc0 : SLOPE * acc0;
    o.y = acc1 >= 0.f ? acc1 : SLOPE * acc1;
    o.z = acc2 >= 0.f ? acc2 : SLOPE * acc2;
    o.w = acc3 >= 0.f ? acc3 : SLOPE * acc3;

    float* optr = out + ((n * C + c) * H + h) * W + wbase;
    __builtin_nontemporal_store(o, (v4f*)optr);
}

extern "C" void kernel_launch(void* const* d_in, const int* in_sizes, int n_in,
                              void* d_out, int out_size, void* d_ws, size_t ws_size,
                              hipStream_t stream) {
    (void)in_sizes; (void)n_in; (void)out_size; (void)d_ws; (void)ws_size;
    const float* feat = (const float*)d_in[0];
    const float* flt  = (const float*)d_in[1];
    float*       out  = (float*)d_out;

    const int blocks = N * C * (H / TILE_H);    // 8192 blocks x 256 threads (8 waves)
    fac_leaky_kernel<<<blocks, 256, 0, stream>>>(feat, flt, out);
}